// CultioGraphNet_19920058319389
// MI455X (gfx1250) — compile-verified
//
#include <hip/hip_runtime.h>
#include <math.h>

typedef float v2f __attribute__((ext_vector_type(2)));
typedef float v8f __attribute__((ext_vector_type(8)));

#define NN_   65536
#define EE_   524288
#define HCATW 229

static __device__ __forceinline__ v8f wmma4(v2f a, v2f b, v8f c) {
  return __builtin_amdgcn_wmma_f32_16x16x4_f32(false, a, false, b, (short)0, c, false, false);
}

// ---------------------------------------------------------------- GEMM (WMMA)
// Out[m, n] = sum_k A[m*ars + k*acs] * B[k, n]  (+ bias[n])
// B row-major K x N, or (bT) B[n*K + k].  N <= 16*NT.
#define GK_KC 288
template <int NT>
__global__ void __launch_bounds__(256) gemm_wmma(
    const float* __restrict__ A, long long ars, long long acs,
    const float* __restrict__ B, int bT,
    const float* __restrict__ bias,
    float* __restrict__ Out, long long ors, long long ocs,
    int M, int K, int N)
{
  __shared__ float sB[GK_KC * 32];
  const int lane = threadIdx.x & 31;
  const int wave = threadIdx.x >> 5;
  const int half = lane >> 4, lm = lane & 15;
  const int tile = blockIdx.x * 8 + wave;
  v8f acc[NT];
#pragma unroll
  for (int nt = 0; nt < NT; ++nt) {
    int col = nt * 16 + lm;
    float bv = (bias && col < N) ? bias[col] : 0.f;
#pragma unroll
    for (int r = 0; r < 8; ++r) acc[nt][r] = bv;
  }
  for (int k0 = 0; k0 < K; k0 += GK_KC) {
    int kc = K - k0; if (kc > GK_KC) kc = GK_KC;
    int kcp = (kc + 3) & ~3;
    for (int i = threadIdx.x; i < kcp * 32; i += 256) {
      int n = i & 31, k = i >> 5;
      float v = 0.f;
      if (k < kc && n < N)
        v = bT ? B[(long long)n * K + k0 + k] : B[(long long)(k0 + k) * N + n];
      sB[k * 32 + n] = v;
    }
    __syncthreads();
    long long row = (long long)tile * 16 + lm;
    bool rok = row < M;
    for (int kk = 0; kk < kcp; kk += 4) {
      int ka = k0 + kk + 2 * half;
      v2f a;
      a.x = (rok && ka     < K) ? A[row * ars + (long long)ka * acs]       : 0.f;
      a.y = (rok && ka + 1 < K) ? A[row * ars + (long long)(ka + 1) * acs] : 0.f;
#pragma unroll
      for (int nt = 0; nt < NT; ++nt) {
        v2f b;
        b.x = sB[(kk + 2 * half) * 32 + nt * 16 + lm];
        b.y = sB[(kk + 2 * half + 1) * 32 + nt * 16 + lm];
        acc[nt] = wmma4(a, b, acc[nt]);
      }
    }
    __syncthreads();
  }
#pragma unroll
  for (int nt = 0; nt < NT; ++nt) {
    int col = nt * 16 + lm;
    if (col >= N) continue;
#pragma unroll
    for (int r = 0; r < 8; ++r) {
      long long row = (long long)tile * 16 + r + 8 * half;
      if (row < M) Out[row * ors + (long long)col * ocs] = acc[nt][r];
    }
  }
}

// ---------------------------------------------------------------- 3x3 conv (implicit GEMM, WMMA)
static __device__ __forceinline__ float conv_fetch(
    int mode, const float* in0, const float* in1, int sOff,
    int Wd, int Hh, int npix, int y0, int x0, int k)
{
  int ci = k / 9;
  int rem = k - ci * 9;
  int dy = rem / 3 - 1;
  int dx = rem - (rem / 3) * 3 - 1;
  int y = y0 + dy, x = x0 + dx;
  if ((unsigned)y >= (unsigned)Hh || (unsigned)x >= (unsigned)Wd) return 0.f;
  int p = y * Wd + x;
  if (mode == 0) return in0[(long long)p * 36 + sOff + ci];
  if (mode == 1) return in0[(long long)ci * npix + p];
  if (ci < 32)   return in0[(long long)ci * npix + p];
  return in1[(long long)(ci - 32) * (npix >> 2) + (y >> 1) * (Wd >> 1) + (x >> 1)];
}

#define CV_KC 144
template <int NT>
__global__ void __launch_bounds__(256) conv3x3_wmma(
    const float* __restrict__ in0, const float* __restrict__ in1,
    int sOff, int mode,
    const float* __restrict__ w, const float* __restrict__ bias,
    float* __restrict__ out, int I, int O, int Wd, int Hh)
{
  __shared__ float sB[CV_KC * 64];
  const int npix = Wd * Hh;
  const int K = I * 9;  // always multiple of 4 here
  const int lane = threadIdx.x & 31, wave = threadIdx.x >> 5;
  const int half = lane >> 4, lm = lane & 15;
  const int tile = blockIdx.x * 8 + wave;
  const int pix = tile * 16 + lm;
  const bool pok = pix < npix;
  const int y0 = pok ? pix / Wd : 0;
  const int x0 = pok ? pix - y0 * Wd : 0;
  v8f acc[NT];
#pragma unroll
  for (int nt = 0; nt < NT; ++nt) {
    int col = nt * 16 + lm;
    float bv = (col < O) ? bias[col] : 0.f;
#pragma unroll
    for (int r = 0; r < 8; ++r) acc[nt][r] = bv;
  }
  for (int k0 = 0; k0 < K; k0 += CV_KC) {
    int kc = K - k0; if (kc > CV_KC) kc = CV_KC;
    for (int i = threadIdx.x; i < kc * 64; i += 256) {
      int n = i & 63, k = i >> 6;
      sB[k * 64 + n] = (n < O) ? w[(long long)n * K + k0 + k] : 0.f;
    }
    __syncthreads();
    for (int kk = 0; kk < kc; kk += 4) {
      int ka = k0 + kk + 2 * half;
      v2f a;
      a.x = pok ? conv_fetch(mode, in0, in1, sOff, Wd, Hh, npix, y0, x0, ka)     : 0.f;
      a.y = pok ? conv_fetch(mode, in0, in1, sOff, Wd, Hh, npix, y0, x0, ka + 1) : 0.f;
#pragma unroll
      for (int nt = 0; nt < NT; ++nt) {
        v2f b;
        b.x = sB[(kk + 2 * half) * 64 + nt * 16 + lm];
        b.y = sB[(kk + 2 * half + 1) * 64 + nt * 16 + lm];
        acc[nt] = wmma4(a, b, acc[nt]);
      }
    }
    __syncthreads();
  }
#pragma unroll
  for (int nt = 0; nt < NT; ++nt) {
    int col = nt * 16 + lm;
    if (col >= O) continue;
#pragma unroll
    for (int r = 0; r < 8; ++r) {
      int prow = tile * 16 + r + 8 * half;
      if (prow < npix) out[(long long)col * npix + prow] = acc[nt][r];
    }
  }
}

// ---------------------------------------------------------------- small utility kernels
__global__ void maxpool2x2(const float* in, float* out, int C, int Wo, int Ho) {
  int idx = blockIdx.x * blockDim.x + threadIdx.x;
  int tot = C * Wo * Ho; if (idx >= tot) return;
  int c = idx / (Wo * Ho); int p = idx - c * (Wo * Ho);
  int y = p / Wo, x = p - y * Wo;
  int Wi = Wo * 2;
  const float* b = in + (long long)c * (Wi * Ho * 2) + (2 * y) * Wi + 2 * x;
  out[idx] = fmaxf(fmaxf(b[0], b[1]), fmaxf(b[Wi], b[Wi + 1]));
}

__global__ void bn_stats(const float* __restrict__ in, long long cstride, long long estride,
                         int count, float* mean, float* var) {
  __shared__ float ss[256], sq[256];
  int c = blockIdx.x;
  const float* p = in + (long long)c * cstride;
  float s = 0.f, q = 0.f;
  for (int i = threadIdx.x; i < count; i += 256) { float v = p[(long long)i * estride]; s += v; q += v * v; }
  ss[threadIdx.x] = s; sq[threadIdx.x] = q;
  __syncthreads();
  for (int o = 128; o > 0; o >>= 1) {
    if ((int)threadIdx.x < o) { ss[threadIdx.x] += ss[threadIdx.x + o]; sq[threadIdx.x] += sq[threadIdx.x + o]; }
    __syncthreads();
  }
  if (threadIdx.x == 0) { float m = ss[0] / count; mean[c] = m; var[c] = sq[0] / count - m * m; }
}

__global__ void bn_apply(float* __restrict__ buf, long long cstride, long long estride,
                         int count, int C, const float* mean, const float* var,
                         const float* g, const float* b, int act) {
  long long idx = (long long)blockIdx.x * blockDim.x + threadIdx.x;
  if (idx >= (long long)count * C) return;
  int c = (int)(idx / count); long long i = idx - (long long)c * count;
  float* p = buf + (long long)c * cstride + i * estride;
  float y = g[c] * (*p - mean[c]) * rsqrtf(var[c] + 1e-5f) + b[c];
  if (act == 1) y = fmaxf(y, 0.f);
  else if (act == 2) y = y > 0.f ? y : 0.1f * expm1f(y);
  *p = y;
}

__global__ void act_strided(float* buf, long long rs, int C, long long M, int act) {
  long long idx = (long long)blockIdx.x * blockDim.x + threadIdx.x;
  if (idx >= M * C) return;
  long long row = idx / C; int c = (int)(idx - row * C);
  float* p = buf + row * rs + c;
  float y = *p;
  if (act == 2) y = y > 0.f ? y : 0.1f * expm1f(y);
  else if (act == 1) y = fmaxf(y, 0.f);
  *p = y;
}

__global__ void copy_strided(const float* src, long long srs, float* dst, long long drs,
                             int C, long long M) {
  long long idx = (long long)blockIdx.x * blockDim.x + threadIdx.x;
  if (idx >= M * C) return;
  long long row = idx / C; int c = (int)(idx - row * C);
  dst[row * drs + c] = src[row * srs + c];
}

__global__ void fill_f32(float* p, float v, long long n) {
  long long i = (long long)blockIdx.x * blockDim.x + threadIdx.x;
  if (i < n) p[i] = v;
}
__global__ void fill_u32(unsigned* p, unsigned v, long long n) {
  long long i = (long long)blockIdx.x * blockDim.x + threadIdx.x;
  if (i < n) p[i] = v;
}

// ---------------------------------------------------------------- TransformerConv edge kernels
__global__ void tc_alpha(const int* __restrict__ ei, const float* __restrict__ ea,
                         const float* __restrict__ q, const float* __restrict__ k,
                         const float* __restrict__ ew, const float* __restrict__ eb,
                         int C, float scale, float* alpha, unsigned* segmax) {
  int e = blockIdx.x * blockDim.x + threadIdx.x;
  if (e >= EE_) return;
  int src = ei[e], dst = ei[EE_ + e];
  float a0 = ea[2 * e], a1 = ea[2 * e + 1];
  float acc = 0.f;
  for (int f = 0; f < C; ++f) {
    float ev = a0 * ew[f] + a1 * ew[C + f] + eb[f];
    acc += q[(long long)dst * C + f] * (k[(long long)src * C + f] + ev);
  }
  acc *= scale;
  alpha[e] = acc;
  unsigned enc = __float_as_uint(acc);
  enc = (enc & 0x80000000u) ? ~enc : (enc | 0x80000000u);
  atomicMax(&segmax[dst], enc);
}

__global__ void tc_expsum(const int* ei, float* alpha, const unsigned* segmax, float* segsum) {
  int e = blockIdx.x * blockDim.x + threadIdx.x;
  if (e >= EE_) return;
  int dst = ei[EE_ + e];
  unsigned u = segmax[dst];
  unsigned bits = (u & 0x80000000u) ? (u & 0x7FFFFFFFu) : ~u;
  float v = expf(alpha[e] - __uint_as_float(bits));
  alpha[e] = v;
  atomicAdd(&segsum[dst], v);
}

__global__ void tc_scatter(const int* ei, const float* ea, const float* v,
                           const float* ew, const float* eb, int C,
                           const float* alpha, const float* segsum,
                           float* out, long long ors) {
  int e = blockIdx.x * blockDim.x + threadIdx.x;
  if (e >= EE_) return;
  int src = ei[e], dst = ei[EE_ + e];
  float wgt = alpha[e] / (segsum[dst] + 1e-16f);
  float a0 = ea[2 * e], a1 = ea[2 * e + 1];
  for (int f = 0; f < C; ++f) {
    float ev = a0 * ew[f] + a1 * ew[C + f] + eb[f];
    atomicAdd(&out[(long long)dst * ors + f], wgt * (v[(long long)src * C + f] + ev));
  }
}

// ---------------------------------------------------------------- GCN kernels
__global__ void gcn_deg(const int* ei, const float* ea, float* deg) {
  int e = blockIdx.x * blockDim.x + threadIdx.x;
  if (e >= EE_) return;
  atomicAdd(&deg[ei[EE_ + e]], ea[2 * e + 1]);
}
__global__ void rsqrt_k(const float* in, float* out, long long n) {
  long long i = (long long)blockIdx.x * blockDim.x + threadIdx.x;
  if (i < n) out[i] = rsqrtf(in[i]);
}
__global__ void gcn_self(const float* xw, const float* dinv, const float* bias, float* z, long long n) {
  long long idx = (long long)blockIdx.x * blockDim.x + threadIdx.x;
  if (idx >= n) return;
  long long row = idx >> 5; int f = (int)(idx & 31);
  float di = dinv[row];
  z[idx] = 2.f * di * di * xw[idx] + bias[f];
}
__global__ void gcn_scatter(const int* ei, const float* ea, const float* xw, const float* dinv, float* z) {
  int e = blockIdx.x * blockDim.x + threadIdx.x;
  if (e >= EE_) return;
  int src = ei[e], dst = ei[EE_ + e];
  float norm = dinv[src] * ea[2 * e + 1] * dinv[dst];
  for (int f = 0; f < 32; ++f)
    atomicAdd(&z[(long long)dst * 32 + f], norm * xw[(long long)src * 32 + f]);
}

// ---------------------------------------------------------------- STAR RNN (WMMA, pipelined layers)
struct StarParams {
  const float* wxk[4]; const float* whk[4]; const float* wxz[4];
  const float* bk[4];  const float* bz[4];
  const float* outw;   const float* outb;
  const float* x;      float* hcat;
};

#define SOFF_WHK 0
#define SOFF_WXK 4096
#define SOFF_WXZ (4096 + 3200)
#define SOFF_BK  (4096 + 3200 + 3200)
#define SOFF_BZ  (SOFF_BK + 128)
#define SOFF_OW  (SOFF_BZ + 128)
#define SOFF_OB  (SOFF_OW + 1024)
#define SW_TOTAL (SOFF_OB + 32)   // 11808 floats

static __device__ __forceinline__ v2f star_ldB(const float* sW, int base, int kk, int nt, int lane) {
  int half = lane >> 4, lm = lane & 15;
  int k = kk + 2 * half;
  v2f b;
  b.x = sW[base + k * 32 + nt * 16 + lm];
  b.y = sW[base + (k + 1) * 32 + nt * 16 + lm];
  return b;
}
static __device__ __forceinline__ v2f star_ldA(const float* st, int kk, int lane) {
  int half = lane >> 4, lm = lane & 15;
  int k = kk + 2 * half;
  v2f a; a.x = st[lm * 33 + k]; a.y = st[lm * 33 + k + 1]; return a;
}
static __device__ __forceinline__ void star_stC(float* st, v8f c, int nt, int lane) {
  int half = lane >> 4, lm = lane & 15;
#pragma unroll
  for (int r = 0; r < 8; ++r) st[(r + 8 * half) * 33 + nt * 16 + lm] = c[r];
}

__global__ void __launch_bounds__(128) star_kernel(StarParams sp) {
  __shared__ float sW[SW_TOTAL];
  __shared__ float sStage[4][2][16 * 33];
  const int tid = threadIdx.x;
  for (int i = tid; i < 4096; i += 128) { int l = i >> 10; sW[SOFF_WHK + i] = sp.whk[l][i & 1023]; }
  for (int i = tid; i < 128; i += 128) {
    int kk = i >> 5, n = i & 31;
    sW[SOFF_WXK + i] = (kk < 3) ? sp.wxk[0][kk * 32 + n] : 0.f;
    sW[SOFF_WXZ + i] = (kk < 3) ? sp.wxz[0][kk * 32 + n] : 0.f;
  }
  for (int i = tid; i < 3072; i += 128) {
    int l = 1 + (i >> 10); int j = i & 1023;
    sW[SOFF_WXK + 128 + i] = sp.wxk[l][j];
    sW[SOFF_WXZ + 128 + i] = sp.wxz[l][j];
  }
  for (int i = tid; i < 128; i += 128) {
    sW[SOFF_BK + i] = sp.bk[i >> 5][i & 31];
    sW[SOFF_BZ + i] = sp.bz[i >> 5][i & 31];
  }
  for (int i = tid; i < 1024; i += 128) sW[SOFF_OW + i] = sp.outw[i];
  for (int i = tid; i < 32; i += 128)   sW[SOFF_OB + i] = sp.outb[i];
  __syncthreads();

  const int lane = tid & 31, wave = tid >> 5;
  const int half = lane >> 4, lm = lane & 15;
  float* Hst = &sStage[wave][0][0];
  float* Xst = &sStage[wave][1][0];
  const long long n0 = ((long long)blockIdx.x * 4 + wave) * 16;

  v8f h[4][2];
#pragma unroll
  for (int l = 0; l < 4; ++l)
#pragma unroll
    for (int nt = 0; nt < 2; ++nt)
#pragma unroll
      for (int r = 0; r < 8; ++r) h[l][nt][r] = 0.f;

  for (int t = 0; t < 12; ++t) {
#pragma unroll
    for (int l = 0; l < 4; ++l) {
      star_stC(Hst, h[l][0], 0, lane);
      star_stC(Hst, h[l][1], 1, lane);
      v8f ak[2], az[2];
      {
        float bk0 = sW[SOFF_BK + l * 32 + lm], bk1 = sW[SOFF_BK + l * 32 + 16 + lm];
        float bz0 = sW[SOFF_BZ + l * 32 + lm], bz1 = sW[SOFF_BZ + l * 32 + 16 + lm];
#pragma unroll
        for (int r = 0; r < 8; ++r) { ak[0][r] = bk0; ak[1][r] = bk1; az[0][r] = bz0; az[1][r] = bz1; }
      }
      for (int kk = 0; kk < 32; kk += 4) {
        v2f a = star_ldA(Hst, kk, lane);
        ak[0] = wmma4(a, star_ldB(sW, SOFF_WHK + l * 1024, kk, 0, lane), ak[0]);
        ak[1] = wmma4(a, star_ldB(sW, SOFF_WHK + l * 1024, kk, 1, lane), ak[1]);
      }
      if (l == 0) {
        int k0 = 2 * half;
        long long row = n0 + lm;
        v2f a;
        a.x = sp.x[row * 36 + (long long)k0 * 12 + t];
        a.y = (k0 + 1 < 3) ? sp.x[row * 36 + (long long)(k0 + 1) * 12 + t] : 0.f;
        ak[0] = wmma4(a, star_ldB(sW, SOFF_WXK, 0, 0, lane), ak[0]);
        ak[1] = wmma4(a, star_ldB(sW, SOFF_WXK, 0, 1, lane), ak[1]);
        az[0] = wmma4(a, star_ldB(sW, SOFF_WXZ, 0, 0, lane), az[0]);
        az[1] = wmma4(a, star_ldB(sW, SOFF_WXZ, 0, 1, lane), az[1]);
      } else {
        int wxkB = SOFF_WXK + 128 + (l - 1) * 1024;
        int wxzB = SOFF_WXZ + 128 + (l - 1) * 1024;
        for (int kk = 0; kk < 32; kk += 4) {
          v2f a = star_ldA(Xst, kk, lane);
          ak[0] = wmma4(a, star_ldB(sW, wxkB, kk, 0, lane), ak[0]);
          ak[1] = wmma4(a, star_ldB(sW, wxkB, kk, 1, lane), ak[1]);
          az[0] = wmma4(a, star_ldB(sW, wxzB, kk, 0, lane), az[0]);
          az[1] = wmma4(a, star_ldB(sW, wxzB, kk, 1, lane), az[1]);
        }
      }
#pragma unroll
      for (int nt = 0; nt < 2; ++nt)
#pragma unroll
        for (int r = 0; r < 8; ++r) {
          float Kg = 1.f / (1.f + expf(-ak[nt][r]));
          float zz = tanhf(az[nt][r]);
          float hc = h[l][nt][r];
          h[l][nt][r] = tanhf(hc + Kg * (zz - hc));
        }
      star_stC(Xst, h[l][0], 0, lane);
      star_stC(Xst, h[l][1], 1, lane);
    }
  }
  star_stC(Hst, h[3][0], 0, lane);
  star_stC(Hst, h[3][1], 1, lane);
  v8f acc[2];
  {
    float b0 = sW[SOFF_OB + lm], b1 = sW[SOFF_OB + 16 + lm];
#pragma unroll
    for (int r = 0; r < 8; ++r) { acc[0][r] = b0; acc[1][r] = b1; }
  }
  for (int kk = 0; kk < 32; kk += 4) {
    v2f a = star_ldA(Hst, kk, lane);
    acc[0] = wmma4(a, star_ldB(sW, SOFF_OW, kk, 0, lane), acc[0]);
    acc[1] = wmma4(a, star_ldB(sW, SOFF_OW, kk, 1, lane), acc[1]);
  }
#pragma unroll
  for (int nt = 0; nt < 2; ++nt)
#pragma unroll
    for (int r = 0; r < 8; ++r) {
      long long row = n0 + r + 8 * half;
      sp.hcat[row * HCATW + 192 + nt * 16 + lm] = acc[nt][r];
    }
}

// ---------------------------------------------------------------- input indices (jax tree order: sorted dict keys)
enum {
  IN_X = 0, IN_EI, IN_EA, IN_BATCH, IN_H, IN_W,
  CLS_BN1_B, CLS_BN1_G, CLS_BN2_B, CLS_BN2_G, CLS_GCN_B, CLS_GCN_W,
  CLS_T_E_B, CLS_T_E_W, CLS_T_K_B, CLS_T_K_W, CLS_T_Q_B, CLS_T_Q_W,
  CLS_T_SKIP_B, CLS_T_SKIP_W, CLS_T_V_B, CLS_T_V_W,
  DIST_BN1_B, DIST_BN1_G, DIST_BN2_B, DIST_BN2_G, DIST_GCN_B, DIST_GCN_W,
  DIST_LIN_B, DIST_LIN_W,
  DIST_T_E_B, DIST_T_E_W, DIST_T_K_B, DIST_T_K_W, DIST_T_Q_B, DIST_T_Q_W,
  DIST_T_SKIP_B, DIST_T_SKIP_W, DIST_T_V_B, DIST_T_V_W,
  EDG_BN1_B, EDG_BN1_G, EDG_BN2_B, EDG_BN2_G, EDG_GCN_B, EDG_GCN_W,
  EDG_T_E_B, EDG_T_E_W, EDG_T_K_B, EDG_T_K_W, EDG_T_Q_B, EDG_T_Q_W,
  EDG_T_SKIP_B, EDG_T_SKIP_W, EDG_T_V_B, EDG_T_V_W,
  B00_BN1_B, B00_BN1_G, B00_BN2_B, B00_BN2_G, B00_C1_B, B00_C1_W, B00_C2_B, B00_C2_W,
  B01_BN1_B, B01_BN1_G, B01_BN2_B, B01_BN2_G, B01_C1_B, B01_C1_W, B01_C2_B, B01_C2_W,
  B10_BN1_B, B10_BN1_G, B10_BN2_B, B10_BN2_G, B10_C1_B, B10_C1_W, B10_C2_B, B10_C2_W,
  FIN_B, FIN_W,
  S0_BK, S0_BZ, S0_WHK, S0_WXK, S0_WXZ,
  S1_BK, S1_BZ, S1_WHK, S1_WXK, S1_WXZ,
  S2_BK, S2_BZ, S2_WHK, S2_WXK, S2_WXZ,
  S3_BK, S3_BZ, S3_WHK, S3_WXK, S3_WXZ,
  STAR_OUT_B, STAR_OUT_W,
  TR_E_B, TR_E_W, TR_K_B, TR_K_W, TR_Q_B, TR_Q_W, TR_SKIP_B, TR_SKIP_W, TR_V_B, TR_V_W
};

extern "C" void kernel_launch(void* const* d_in, const int* in_sizes, int n_in,
                              void* d_out, int out_size, void* d_ws, size_t ws_size,
                              hipStream_t stream)
{
  (void)in_sizes; (void)n_in; (void)out_size; (void)ws_size;
  auto F = [&](int i) { return (const float*)d_in[i]; };
  const float* x  = F(IN_X);
  const int*   ei = (const int*)d_in[IN_EI];
  const float* ea = F(IN_EA);
  float* ws = (float*)d_ws;
  float* outp = (float*)d_out;

  size_t o = 0;
  float* hcat = ws + o; o += (size_t)NN_ * HCATW;
  float* pl0  = ws + o; o += (size_t)32 * NN_;
  float* pl1  = ws + o; o += (size_t)32 * NN_;
  float* pl2  = ws + o; o += (size_t)64 * 16384;
  float* pl3  = ws + o; o += (size_t)64 * 16384;
  float* qb   = ws + o; o += (size_t)NN_ * 32;
  float* kb   = ws + o; o += (size_t)NN_ * 32;
  float* vb   = ws + o; o += (size_t)NN_ * 32;
  float* xw   = ws + o; o += (size_t)NN_ * 32;
  float* z1   = ws + o; o += (size_t)NN_ * 32;
  float* z2   = ws + o; o += (size_t)NN_ * 32;
  float* alpha = ws + o; o += (size_t)EE_;
  unsigned* segmax = (unsigned*)(ws + o); o += NN_;
  float* segsum = ws + o; o += NN_;
  float* deg    = ws + o; o += NN_;
  float* dinv   = ws + o; o += NN_;
  float* meanb  = ws + o; o += 64;
  float* varb   = ws + o; o += 64;

  auto gemm = [&](const float* A, long long ars, long long acs,
                  const float* B, int bT, const float* bias,
                  float* Out, long long ors, long long ocs, int M, int K, int N) {
    int tiles = (M + 15) / 16;
    int grid = (tiles + 7) / 8;
    if (N > 16)
      gemm_wmma<2><<<grid, 256, 0, stream>>>(A, ars, acs, B, bT, bias, Out, ors, ocs, M, K, N);
    else
      gemm_wmma<1><<<grid, 256, 0, stream>>>(A, ars, acs, B, bT, bias, Out, ors, ocs, M, K, N);
  };
  auto conv = [&](const float* in0, const float* in1, int sOff, int mode,
                  const float* w, const float* b, float* out0, int I, int O, int Wd, int Hh) {
    int tiles = (Wd * Hh + 15) / 16;
    int grid = (tiles + 7) / 8;
    if (O > 32)
      conv3x3_wmma<4><<<grid, 256, 0, stream>>>(in0, in1, sOff, mode, w, b, out0, I, O, Wd, Hh);
    else
      conv3x3_wmma<2><<<grid, 256, 0, stream>>>(in0, in1, sOff, mode, w, b, out0, I, O, Wd, Hh);
  };
  auto bn2d = [&](float* buf, int C, int npix, const float* g, const float* b) {
    bn_stats<<<C, 256, 0, stream>>>(buf, npix, 1, npix, meanb, varb);
    bn_apply<<<(int)(((long long)C * npix + 255) / 256), 256, 0, stream>>>(
        buf, npix, 1, npix, C, meanb, varb, g, b, 1);
  };

  // ---------------- transformer stream (tstream -> hcat cols [0,96))
  for (int s = 0; s < 3; ++s) {
    const float* xs = x + s * 12;
    gemm(xs, 36, 1, F(TR_Q_W), 0, F(TR_Q_B), qb, 32, 1, NN_, 12, 32);
    gemm(xs, 36, 1, F(TR_K_W), 0, F(TR_K_B), kb, 32, 1, NN_, 12, 32);
    gemm(xs, 36, 1, F(TR_V_W), 0, F(TR_V_B), vb, 32, 1, NN_, 12, 32);
    gemm(xs, 36, 1, F(TR_SKIP_W), 0, F(TR_SKIP_B), hcat + s * 32, HCATW, 1, NN_, 12, 32);
    fill_u32<<<NN_ / 256, 256, 0, stream>>>(segmax, 0u, NN_);
    tc_alpha<<<EE_ / 256, 256, 0, stream>>>(ei, ea, qb, kb, F(TR_E_W), F(TR_E_B),
                                            32, 1.f / sqrtf(32.f), alpha, segmax);
    fill_f32<<<NN_ / 256, 256, 0, stream>>>(segsum, 0.f, NN_);
    tc_expsum<<<EE_ / 256, 256, 0, stream>>>(ei, alpha, segmax, segsum);
    tc_scatter<<<EE_ / 256, 256, 0, stream>>>(ei, ea, vb, F(TR_E_W), F(TR_E_B), 32,
                                              alpha, segsum, hcat + s * 32, HCATW);
    act_strided<<<(int)(((long long)NN_ * 32 + 255) / 256), 256, 0, stream>>>(
        hcat + s * 32, HCATW, 32, NN_, 2);
  }

  // ---------------- nunet stream (nstream -> hcat cols [96,192))
  for (int s = 0; s < 3; ++s) {
    conv(x, nullptr, s * 12, 0, F(B00_C1_W), F(B00_C1_B), pl0, 12, 32, 256, 256);
    bn2d(pl0, 32, 65536, F(B00_BN1_G), F(B00_BN1_B));
    conv(pl0, nullptr, 0, 1, F(B00_C2_W), F(B00_C2_B), pl1, 32, 32, 256, 256);
    bn2d(pl1, 32, 65536, F(B00_BN2_G), F(B00_BN2_B));      // pl1 = x00
    maxpool2x2<<<(32 * 16384 + 255) / 256, 256, 0, stream>>>(pl1, pl2, 32, 128, 128);
    conv(pl2, nullptr, 0, 1, F(B10_C1_W), F(B10_C1_B), pl3, 32, 64, 128, 128);
    bn2d(pl3, 64, 16384, F(B10_BN1_G), F(B10_BN1_B));
    conv(pl3, nullptr, 0, 1, F(B10_C2_W), F(B10_C2_B), pl2, 64, 64, 128, 128);
    bn2d(pl2, 64, 16384, F(B10_BN2_G), F(B10_BN2_B));      // pl2 = x10
    conv(pl1, pl2, 0, 2, F(B01_C1_W), F(B01_C1_B), pl0, 96, 32, 256, 256);
    bn2d(pl0, 32, 65536, F(B01_BN1_G), F(B01_BN1_B));
    conv(pl0, nullptr, 0, 1, F(B01_C2_W), F(B01_C2_B), pl1, 32, 32, 256, 256);
    bn2d(pl1, 32, 65536, F(B01_BN2_G), F(B01_BN2_B));      // pl1 = x01
    gemm(pl1, 1, NN_, F(FIN_W), 1, F(FIN_B), hcat + 96 + s * 32, HCATW, 1, NN_, 32, 32);
  }

  // ---------------- STAR RNN (sstream -> hcat cols [192,224))
  {
    StarParams sp;
    const int WXK[4] = {S0_WXK, S1_WXK, S2_WXK, S3_WXK};
    const int WHK[4] = {S0_WHK, S1_WHK, S2_WHK, S3_WHK};
    const int WXZ[4] = {S0_WXZ, S1_WXZ, S2_WXZ, S3_WXZ};
    const int BK[4]  = {S0_BK, S1_BK, S2_BK, S3_BK};
    const int BZ[4]  = {S0_BZ, S1_BZ, S2_BZ, S3_BZ};
    for (int l = 0; l < 4; ++l) {
      sp.wxk[l] = F(WXK[l]); sp.whk[l] = F(WHK[l]); sp.wxz[l] = F(WXZ[l]);
      sp.bk[l] = F(BK[l]);   sp.bz[l] = F(BZ[l]);
    }
    sp.outw = F(STAR_OUT_W); sp.outb = F(STAR_OUT_B);
    sp.x = x; sp.hcat = hcat;
    star_kernel<<<1024, 128, 0, stream>>>(sp);
  }

  // ---------------- GCN degree normalization (shared by all heads)
  fill_f32<<<NN_ / 256, 256, 0, stream>>>(deg, 2.f, NN_);
  gcn_deg<<<EE_ / 256, 256, 0, stream>>>(ei, ea, deg);
  rsqrt_k<<<NN_ / 256, 256, 0, stream>>>(deg, dinv, NN_);

  auto head = [&](int Kin, const float* gw, const float* gb,
                  const float* bn1g, const float* bn1b,
                  const float* tew, const float* teb,
                  const float* tkw, const float* tkb,
                  const float* tqw, const float* tqb,
                  const float* tsw, const float* tsb,
                  const float* tvw, const float* tvb,
                  const float* bn2g, const float* bn2b, int C2) {
    gemm(hcat, HCATW, 1, gw, 0, nullptr, xw, 32, 1, NN_, Kin, 32);
    gcn_self<<<(int)(((long long)NN_ * 32 + 255) / 256), 256, 0, stream>>>(
        xw, dinv, gb, z1, (long long)NN_ * 32);
    gcn_scatter<<<EE_ / 256, 256, 0, stream>>>(ei, ea, xw, dinv, z1);
    bn_stats<<<32, 256, 0, stream>>>(z1, 1, 32, NN_, meanb, varb);
    bn_apply<<<(int)(((long long)NN_ * 32 + 255) / 256), 256, 0, stream>>>(
        z1, 1, 32, NN_, 32, meanb, varb, bn1g, bn1b, 0);
    gemm(z1, 32, 1, tqw, 0, tqb, qb, C2, 1, NN_, 32, C2);
    gemm(z1, 32, 1, tkw, 0, tkb, kb, C2, 1, NN_, 32, C2);
    gemm(z1, 32, 1, tvw, 0, tvb, vb, C2, 1, NN_, 32, C2);
    gemm(z1, 32, 1, tsw, 0, tsb, z2, C2, 1, NN_, 32, C2);
    fill_u32<<<NN_ / 256, 256, 0, stream>>>(segmax, 0u, NN_);
    tc_alpha<<<EE_ / 256, 256, 0, stream>>>(ei, ea, qb, kb, tew, teb,
                                            C2, 1.f / sqrtf((float)C2), alpha, segmax);
    fill_f32<<<NN_ / 256, 256, 0, stream>>>(segsum, 0.f, NN_);
    tc_expsum<<<EE_ / 256, 256, 0, stream>>>(ei, alpha, segmax, segsum);
    tc_scatter<<<EE_ / 256, 256, 0, stream>>>(ei, ea, vb, tew, teb, C2, alpha, segsum, z2, C2);
    bn_stats<<<C2, 256, 0, stream>>>(z2, 1, C2, NN_, meanb, varb);
    bn_apply<<<(int)(((long long)NN_ * C2 + 255) / 256), 256, 0, stream>>>(
        z2, 1, C2, NN_, C2, meanb, varb, bn2g, bn2b, 2);
  };

  // dist head -> ld (N x 3)
  head(224, F(DIST_GCN_W), F(DIST_GCN_B), F(DIST_BN1_G), F(DIST_BN1_B),
       F(DIST_T_E_W), F(DIST_T_E_B), F(DIST_T_K_W), F(DIST_T_K_B),
       F(DIST_T_Q_W), F(DIST_T_Q_B), F(DIST_T_SKIP_W), F(DIST_T_SKIP_B),
       F(DIST_T_V_W), F(DIST_T_V_B), F(DIST_BN2_G), F(DIST_BN2_B), 32);
  gemm(z2, 32, 1, F(DIST_LIN_W), 0, F(DIST_LIN_B), hcat + 224, HCATW, 1, NN_, 32, 3);
  copy_strided<<<(NN_ * 3 + 255) / 256, 256, 0, stream>>>(hcat + 224, HCATW, outp, 3, 3, NN_);

  // edge head -> le (N x 2)
  head(227, F(EDG_GCN_W), F(EDG_GCN_B), F(EDG_BN1_G), F(EDG_BN1_B),
       F(EDG_T_E_W), F(EDG_T_E_B), F(EDG_T_K_W), F(EDG_T_K_B),
       F(EDG_T_Q_W), F(EDG_T_Q_B), F(EDG_T_SKIP_W), F(EDG_T_SKIP_B),
       F(EDG_T_V_W), F(EDG_T_V_B), F(EDG_BN2_G), F(EDG_BN2_B), 2);
  copy_strided<<<(NN_ * 2 + 255) / 256, 256, 0, stream>>>(z2, 2, hcat + 227, HCATW, 2, NN_);
  copy_strided<<<(NN_ * 2 + 255) / 256, 256, 0, stream>>>(z2, 2, outp + (size_t)NN_ * 3, 2, 2, NN_);

  // cls head -> lc (N x 2)
  head(229, F(CLS_GCN_W), F(CLS_GCN_B), F(CLS_BN1_G), F(CLS_BN1_B),
       F(CLS_T_E_W), F(CLS_T_E_B), F(CLS_T_K_W), F(CLS_T_K_B),
       F(CLS_T_Q_W), F(CLS_T_Q_B), F(CLS_T_SKIP_W), F(CLS_T_SKIP_B),
       F(CLS_T_V_W), F(CLS_T_V_B), F(CLS_BN2_G), F(CLS_BN2_B), 2);
  copy_strided<<<(NN_ * 2 + 255) / 256, 256, 0, stream>>>(z2, 2, outp + (size_t)NN_ * 5, 2, 2, NN_);
}